// RandomEqualize_81612968559296
// MI455X (gfx1250) — compile-verified
//
#include <hip/hip_runtime.h>
#include <hip/hip_bf16.h>

typedef __attribute__((ext_vector_type(2))) float v2f;
typedef __attribute__((ext_vector_type(8))) float v8f;

#define PIX   (512 * 512)   // pixels per channel (H*W from reference)
#define NBINS 256
#define HB    8             // histogram blocks per channel
#define AB    16            // apply blocks per channel

// ---------------------------------------------------------------------------
// Kernel 1: per-channel 256-bin histogram.
// 8 blocks per channel, 256 threads (8 waves). Per-wave LDS sub-histograms
// (no inter-wave LDS contention), reduced and flushed with global atomics.
// ---------------------------------------------------------------------------
__global__ void eq_hist_kernel(const float* __restrict__ img,
                               unsigned int* __restrict__ ghist) {
    __shared__ unsigned int sh[8][NBINS];
    const int tid  = threadIdx.x;
    const int wave = tid >> 5;

    for (int i = tid; i < 8 * NBINS; i += blockDim.x)
        (&sh[0][0])[i] = 0u;
    __syncthreads();

    const int ch    = blockIdx.x / HB;
    const int chunk = blockIdx.x % HB;
    const int pxPerBlock = PIX / HB;                 // 32768
    const float4* src = (const float4*)(img + (size_t)ch * PIX
                                            + (size_t)chunk * pxPerBlock);
    const int n4 = pxPerBlock / 4;                   // 8192

    for (int i = tid; i < n4; i += blockDim.x) {
        __builtin_prefetch(&src[i + 2 * blockDim.x], 0, 0);  // global_prefetch_b8
        float4 f = src[i];
        atomicAdd(&sh[wave][(int)f.x], 1u);
        atomicAdd(&sh[wave][(int)f.y], 1u);
        atomicAdd(&sh[wave][(int)f.z], 1u);
        atomicAdd(&sh[wave][(int)f.w], 1u);
    }
    __syncthreads();

    for (int b = tid; b < NBINS; b += blockDim.x) {
        unsigned int s = 0;
        #pragma unroll
        for (int w = 0; w < 8; ++w) s += sh[w][b];
        if (s) atomicAdd(&ghist[ch * NBINS + b], s);
    }
}

// ---------------------------------------------------------------------------
// Kernel 2: per-channel LUT. One wave32 per block (EXEC all-1s => WMMA legal).
// 256-bin inclusive cumsum done with v_wmma_f32_16x16x4_f32 (exact: all
// values are integers <= 262144 < 2^24).
//   H[r][c] = hist[16r+c]
//   P  = H * U        (U[k][c] = k<=c)        -> within-row prefix
//   cs = L * W + P    (L[r][k] = k<r, W[k][*] = rowsum T[k] = P[k][15])
// Assumed f32 A-layout (ISA 7.12.2): lane<16 holds K=0,1 ; lane>=16 holds
// K=2,3 of row (lane&15); B mirrored (rows K across the two half-waves).
// LUT folds the step==0 pass-through as the identity mapping.
// ---------------------------------------------------------------------------
__global__ void eq_lut_kernel(const unsigned int* __restrict__ ghist,
                              float* __restrict__ glut) {
    __shared__ float h[NBINS];
    __shared__ float tvec[16];
    __shared__ float csarr[NBINS];
    __shared__ int   s_step;

    const int lane = threadIdx.x;      // 0..31, single wave
    const int ch   = blockIdx.x;

    for (int i = lane; i < NBINS; i += 32)
        h[i] = (float)ghist[ch * NBINS + i];
    __syncthreads();

    const int row  = lane & 15;                // M row (A) / N col (B)
    const int koff = (lane < 16) ? 0 : 2;      // K sub-offset for this half-wave

    // ---- P = H x U ----
    v8f acc = {};
    #pragma unroll
    for (int j = 0; j < 4; ++j) {
        const int k0 = 4 * j + koff;
        v2f a; a.x = h[row * 16 + k0];
               a.y = h[row * 16 + k0 + 1];
        v2f b; b.x = (k0     <= row) ? 1.0f : 0.0f;
               b.y = (k0 + 1 <= row) ? 1.0f : 0.0f;
        acc = __builtin_amdgcn_wmma_f32_16x16x4_f32(false, a, false, b,
                                                    (short)0, acc, false, false);
    }

    // Extract row totals T[k] = P[k][15] (D layout: VGPR i, lanes0-15 -> M=i,
    // N=lane ; lanes16-31 -> M=8+i, N=lane-16).
    if (lane == 15) {
        for (int i = 0; i < 8; ++i) tvec[i] = acc[i];
    }
    if (lane == 31) {
        for (int i = 0; i < 8; ++i) tvec[8 + i] = acc[i];
    }
    __syncthreads();

    // ---- cs = Lstrict x W + P ----
    #pragma unroll
    for (int j = 0; j < 4; ++j) {
        const int k0 = 4 * j + koff;
        v2f a; a.x = (k0     < row) ? 1.0f : 0.0f;
               a.y = (k0 + 1 < row) ? 1.0f : 0.0f;
        v2f b; b.x = tvec[k0];
               b.y = tvec[k0 + 1];
        acc = __builtin_amdgcn_wmma_f32_16x16x4_f32(false, a, false, b,
                                                    (short)0, acc, false, false);
    }

    // Scatter cs (D layout) back to LDS as a flat 256-entry array.
    #pragma unroll
    for (int i = 0; i < 8; ++i) {
        const int m = (lane < 16) ? i : (8 + i);
        const int n = lane & 15;
        csarr[m * 16 + n] = acc[i];
    }
    __syncthreads();

    if (lane == 0) {
        int last = NBINS - 1;
        while (last > 0 && h[last] == 0.0f) --last;
        const int lastval = (int)h[last];
        s_step = (PIX - lastval) / 255;
    }
    __syncthreads();

    const int step = s_step;
    for (int i = lane; i < NBINS; i += 32) {
        float outv;
        if (step == 0) {
            outv = (float)i;                       // pass-through as identity LUT
        } else {
            const int cprev = (i == 0) ? 0 : (int)csarr[i - 1];
            int v = (cprev + (step >> 1)) / step;
            v = (v < 0) ? 0 : ((v > 255) ? 255 : v);
            outv = (float)v;
        }
        glut[ch * NBINS + i] = outv;
    }
}

// ---------------------------------------------------------------------------
// Kernel 3: apply LUT. 16 blocks per channel, LUT staged in LDS, B128 streams.
// ---------------------------------------------------------------------------
__global__ void eq_apply_kernel(const float* __restrict__ img,
                                const float* __restrict__ glut,
                                float* __restrict__ out) {
    __shared__ float lut[NBINS];
    const int tid   = threadIdx.x;
    const int ch    = blockIdx.x / AB;
    const int chunk = blockIdx.x % AB;

    for (int i = tid; i < NBINS; i += blockDim.x)
        lut[i] = glut[ch * NBINS + i];
    __syncthreads();

    const int pxPerBlock = PIX / AB;                 // 16384
    const size_t base = (size_t)ch * PIX + (size_t)chunk * pxPerBlock;
    const float4* src = (const float4*)(img + base);
    float4*       dst = (float4*)(out + base);
    const int n4 = pxPerBlock / 4;                   // 4096

    for (int i = tid; i < n4; i += blockDim.x) {
        __builtin_prefetch(&src[i + 2 * blockDim.x], 0, 0);  // global_prefetch_b8
        float4 f = src[i];
        float4 o;
        o.x = lut[(int)f.x];
        o.y = lut[(int)f.y];
        o.z = lut[(int)f.z];
        o.w = lut[(int)f.w];
        dst[i] = o;
    }
}

// ---------------------------------------------------------------------------
extern "C" void kernel_launch(void* const* d_in, const int* in_sizes, int n_in,
                              void* d_out, int out_size, void* d_ws, size_t ws_size,
                              hipStream_t stream) {
    const float* img = (const float*)d_in[0];
    float* out = (float*)d_out;

    const int total = in_sizes[0];          // 64*3*512*512
    const int nch   = total / PIX;          // 192 channels

    unsigned int* ghist = (unsigned int*)d_ws;
    float* glut = (float*)((char*)d_ws + (size_t)nch * NBINS * sizeof(unsigned int));

    (void)hipMemsetAsync(ghist, 0, (size_t)nch * NBINS * sizeof(unsigned int), stream);

    eq_hist_kernel <<<nch * HB, 256, 0, stream>>>(img, ghist);
    eq_lut_kernel  <<<nch,      32,  0, stream>>>(ghist, glut);
    eq_apply_kernel<<<nch * AB, 256, 0, stream>>>(img, glut, out);
}